// SpatialAttentionModule_50251117363781
// MI455X (gfx1250) — compile-verified
//
#include <hip/hip_runtime.h>
#include <hip/hip_bf16.h>
#include <math.h>

typedef float v2f __attribute__((ext_vector_type(2)));
typedef float v8f __attribute__((ext_vector_type(8)));

// ---------------------------------------------------------------------------
// Kernel 1: q/k/v projection via V_WMMA_F32_16X16X4_F32.
// One wave per 16-spatial tile. A = x[16 rows x 4 channels] chunk, B[4x16] has
// col0=q_w, col1=k_w, col2=v_w. Chain 16 WMMAs over the 64 channels.
// A layout (f32 16x4): lane L, row M=L&15; a.x -> K=2*(L>>4), a.y -> K=2*(L>>4)+1
// B layout (f32 4x16): lane L, col N=L&15; b.x -> K=2*(L>>4), b.y -> K=2*(L>>4)+1
// D layout (f32 16x16): VGPR r, lane L: M = r + 8*(L>>4), N = L&15
// ---------------------------------------------------------------------------
__global__ __launch_bounds__(128) void qkv_wmma_kernel(
    const float* __restrict__ x,
    const float* __restrict__ qw, const float* __restrict__ kw,
    const float* __restrict__ vw,
    float* __restrict__ q, float* __restrict__ k, float* __restrict__ v)
{
    const int wave  = threadIdx.x >> 5;
    const int lane  = threadIdx.x & 31;
    const int tile  = blockIdx.x * 4 + wave;          // 1024 tiles total
    const int batch = tile >> 8;                      // 256 tiles per batch
    const int sbase = (tile & 255) << 4;              // 16 spatial rows / tile
    const int row   = lane & 15;
    const int half  = lane >> 4;
    const int col   = row;                            // D column owned by lane

    const float* xb = x + (size_t)batch * 64 * 4096;

    v8f acc = {};
    #pragma unroll
    for (int c16 = 0; c16 < 16; ++c16) {
        const int kg = c16 * 4 + 2 * half;            // channel for a.x / b.x
        v2f a, b;
        a.x = xb[(size_t)kg       * 4096 + sbase + row];
        a.y = xb[(size_t)(kg + 1) * 4096 + sbase + row];
        b.x = (col == 0) ? qw[kg]     : (col == 1) ? kw[kg]     : (col == 2) ? vw[kg]     : 0.0f;
        b.y = (col == 0) ? qw[kg + 1] : (col == 1) ? kw[kg + 1] : (col == 2) ? vw[kg + 1] : 0.0f;
        acc = __builtin_amdgcn_wmma_f32_16x16x4_f32(false, a, false, b,
                                                    (short)0, acc, false, false);
    }

    if (col < 3) {
        float* dst = (col == 0) ? q : (col == 1) ? k : v;
        const int base = batch * 4096 + sbase + 8 * half;
        #pragma unroll
        for (int r = 0; r < 8; ++r) dst[base + r] = acc[r];
    }
}

// ---------------------------------------------------------------------------
// Kernel 2: per-batch max/min of k (for the exact softmax row max:
// m_i = q_i*kmax if q_i>=0 else q_i*kmin, since scores are rank-1).
// ---------------------------------------------------------------------------
__global__ __launch_bounds__(256) void kminmax_kernel(
    const float* __restrict__ k, float* __restrict__ kmax, float* __restrict__ kmin)
{
    __shared__ float smax[256];
    __shared__ float smin[256];
    const int b = blockIdx.x;
    const int t = threadIdx.x;
    const float* kb = k + b * 4096;
    float mx = -INFINITY, mn = INFINITY;
    for (int i = t; i < 4096; i += 256) {
        const float val = kb[i];
        mx = fmaxf(mx, val);
        mn = fminf(mn, val);
    }
    smax[t] = mx; smin[t] = mn;
    __syncthreads();
    for (int s = 128; s > 0; s >>= 1) {
        if (t < s) {
            smax[t] = fmaxf(smax[t], smax[t + s]);
            smin[t] = fminf(smin[t], smin[t + s]);
        }
        __syncthreads();
    }
    if (t == 0) { kmax[b] = smax[0]; kmin[b] = smin[0]; }
}

// ---------------------------------------------------------------------------
// Kernel 3: fused streaming softmax-attention. k,v staged in LDS.
// Per wave: 16 rows i. A tile = exp2(q2*k_j - m2) (native v_exp_f32),
// B col0 = v_j, col1 = 1.0 -> one WMMA chain accumulates numerator (col 0)
// and denominator (col 1) simultaneously. Never materializes 4096x4096.
// ---------------------------------------------------------------------------
__global__ __launch_bounds__(256) void attn_wmma_kernel(
    const float* __restrict__ q, const float* __restrict__ k,
    const float* __restrict__ v, const float* __restrict__ kmax,
    const float* __restrict__ kmin, float* __restrict__ attn)
{
    __shared__ float sk[4096];
    __shared__ float sv[4096];

    const int batch = blockIdx.x >> 5;                // 32 blocks per batch
    const int blk   = blockIdx.x & 31;
    const int tid   = threadIdx.x;

    const float* kb = k + batch * 4096;
    const float* vb = v + batch * 4096;
    for (int i = tid; i < 4096; i += 256) { sk[i] = kb[i]; sv[i] = vb[i]; }
    __syncthreads();

    const int wave  = tid >> 5;
    const int lane  = tid & 31;
    const int row   = lane & 15;
    const int half  = lane >> 4;
    const int col   = row;
    const int itile = blk * 8 + wave;                 // 0..255
    const int i     = itile * 16 + row;

    const float qi = q[batch * 4096 + i];
    const float LOG2E = 1.4426950408889634f;
    const float q2 = qi * LOG2E;
    const float m2 = ((qi >= 0.0f) ? qi * kmax[batch] : qi * kmin[batch]) * LOG2E;

    v8f acc = {};
    #pragma unroll 4
    for (int jb = 0; jb < 4096; jb += 4) {
        const int j0 = jb + 2 * half;
        v2f a, b;
        a.x = exp2f(q2 * sk[j0]     - m2);
        a.y = exp2f(q2 * sk[j0 + 1] - m2);
        b.x = (col == 0) ? sv[j0]     : (col == 1) ? 1.0f : 0.0f;
        b.y = (col == 0) ? sv[j0 + 1] : (col == 1) ? 1.0f : 0.0f;
        acc = __builtin_amdgcn_wmma_f32_16x16x4_f32(false, a, false, b,
                                                    (short)0, acc, false, false);
    }

    // Numerator lives in column 0 (lanes 0/16), denominator in column 1 (lanes 1/17).
    const int denLane = half * 16 + 1;
    #pragma unroll
    for (int r = 0; r < 8; ++r) {
        const float num = acc[r];
        const float den = __shfl(acc[r], denLane, 32);
        if (col == 0) {
            attn[batch * 4096 + itile * 16 + 8 * half + r] = num / den;
        }
    }
}

// ---------------------------------------------------------------------------
// Kernel 4: 7x7 reflect-pad "valid" conv (cross-correlation, matching
// jax.lax.conv_general_dilated). One thread per output pixel.
// reflect (no edge repeat): idx<0 -> -idx ; idx>63 -> 126-idx
// ---------------------------------------------------------------------------
__global__ __launch_bounds__(256) void conv7_reflect_kernel(
    const float* __restrict__ attn, const float* __restrict__ tw,
    float* __restrict__ out)
{
    const int idx = blockIdx.x * 256 + threadIdx.x;   // 16384 pixels
    const int b = idx >> 12;
    const int s = idx & 4095;
    const int y = s >> 6;
    const int x = s & 63;
    const float* ab = attn + b * 4096;

    float acc = 0.0f;
    #pragma unroll
    for (int ky = 0; ky < 7; ++ky) {
        int yy = y + ky - 3;
        yy = (yy < 0) ? -yy : ((yy > 63) ? 126 - yy : yy);
        #pragma unroll
        for (int kx = 0; kx < 7; ++kx) {
            int xx = x + kx - 3;
            xx = (xx < 0) ? -xx : ((xx > 63) ? 126 - xx : xx);
            acc = fmaf(tw[ky * 7 + kx], ab[yy * 64 + xx], acc);
        }
    }
    out[idx] = acc;
}

// ---------------------------------------------------------------------------
extern "C" void kernel_launch(void* const* d_in, const int* in_sizes, int n_in,
                              void* d_out, int out_size, void* d_ws, size_t ws_size,
                              hipStream_t stream) {
    const float* x  = (const float*)d_in[0];   // [4,64,64,64]
    const float* qw = (const float*)d_in[1];   // [64]
    const float* kw = (const float*)d_in[2];   // [64]
    const float* vw = (const float*)d_in[3];   // [64]
    const float* tw = (const float*)d_in[4];   // [1,1,7,7]

    float* wsf   = (float*)d_ws;
    float* qbuf  = wsf;                 // 4*4096
    float* kbuf  = wsf + 16384;         // 4*4096
    float* vbuf  = wsf + 32768;         // 4*4096
    float* abuf  = wsf + 49152;         // 4*4096
    float* kmaxb = wsf + 65536;         // 4
    float* kminb = wsf + 65540;         // 4

    // 1) q,k,v projections (WMMA f32): 1024 tiles, 4 waves/block
    qkv_wmma_kernel<<<256, 128, 0, stream>>>(x, qw, kw, vw, qbuf, kbuf, vbuf);
    // 2) per-batch k range for exact softmax max
    kminmax_kernel<<<4, 256, 0, stream>>>(kbuf, kmaxb, kminb);
    // 3) fused rank-1 softmax attention (WMMA f32 + LDS-staged k/v)
    attn_wmma_kernel<<<128, 256, 0, stream>>>(qbuf, kbuf, vbuf, kmaxb, kminb, abuf);
    // 4) 7x7 reflect conv
    conv7_reflect_kernel<<<64, 256, 0, stream>>>(abuf, tw, (float*)d_out);
}